// LocalEquivScoreModule_11553462026624
// MI455X (gfx1250) — compile-verified
//
#include <hip/hip_runtime.h>
#include <cmath>

#define B_N 4
#define C_N 3
#define H_N 32
#define W_N 32
#define HW (H_N * W_N)            // 1024
#define M_TOTAL (B_N * HW)        // 4096
#define KDIM 27
#define KPAD 28
#define NIMG 8
#define HO 30
#define WO 30
#define P_STEP (NIMG * HO * WO)   // 7200
#define STEPS 4
#define P_TOTAL (STEPS * P_STEP)  // 28800
#define NTILES (P_TOTAL / 16)     // 1800
#define WAVES 4
#define TILES_PER_WAVE (NTILES / WAVES) // 450

typedef float v2f __attribute__((ext_vector_type(2)));
typedef float v8f __attribute__((ext_vector_type(8)));

// ---------------------------------------------------------------------------
// Build pre-paired x-patch matrix XA[14][M_TOTAL] (v2f) + xnorms.
// Pair p = 2q+hi holds K elements (4q+2hi, 4q+2hi+1); k = c*9+i*3+j; k=27 -> 0.
// Wrap padding on H/W.
// ---------------------------------------------------------------------------
__global__ void build_x_kernel(const float* __restrict__ x,
                               v2f* __restrict__ XA,
                               float* __restrict__ xnorms) {
  int Mi = blockIdx.x * blockDim.x + threadIdx.x;
  if (Mi >= M_TOTAL) return;
  int b = Mi >> 10;
  int hw = Mi & (HW - 1);
  int h = hw >> 5;
  int w = hw & 31;
  float vals[KPAD];
  float nrm = 0.0f;
#pragma unroll
  for (int c = 0; c < C_N; ++c) {
#pragma unroll
    for (int i = 0; i < 3; ++i) {
#pragma unroll
      for (int j = 0; j < 3; ++j) {
        int hh = (h + i - 1 + H_N) & (H_N - 1);  // wrap pad
        int ww = (w + j - 1 + W_N) & (W_N - 1);
        float v = x[(b * C_N + c) * HW + hh * W_N + ww];
        vals[c * 9 + i * 3 + j] = v;
        nrm += v * v;
      }
    }
  }
  vals[KDIM] = 0.0f;  // K pad
#pragma unroll
  for (int p = 0; p < 14; ++p) {
    v2f pr;
    pr.x = vals[2 * p];
    pr.y = vals[2 * p + 1];
    XA[p * M_TOTAL + Mi] = pr;
  }
  xnorms[Mi] = nrm;
}

// ---------------------------------------------------------------------------
// Build pre-paired patch matrix PB[14][P_TOTAL] (v2f) for all 4 steps, plus
// PNC[P_TOTAL] = float4(pnorm, center_c0, center_c1, center_c2).
// g = s*7200 + (n*30+y)*30 + xx ; patch[k] = images[s][n][c][y+i][xx+j]
// ---------------------------------------------------------------------------
__global__ void build_p_kernel(const float* __restrict__ images,
                               v2f* __restrict__ PB,
                               float4* __restrict__ PNC) {
  int g = blockIdx.x * blockDim.x + threadIdx.x;
  if (g >= P_TOTAL) return;
  int s = g / P_STEP;
  int rem = g - s * P_STEP;
  int n = rem / (HO * WO);
  int rem2 = rem - n * (HO * WO);
  int y = rem2 / WO;
  int xx = rem2 - y * WO;
  float vals[KPAD];
  float nrm = 0.0f;
#pragma unroll
  for (int c = 0; c < C_N; ++c) {
#pragma unroll
    for (int i = 0; i < 3; ++i) {
#pragma unroll
      for (int j = 0; j < 3; ++j) {
        float v = images[(((s * NIMG + n) * C_N + c) * H_N + (y + i)) * W_N + (xx + j)];
        vals[c * 9 + i * 3 + j] = v;
        nrm += v * v;
      }
    }
  }
  vals[KDIM] = 0.0f;
#pragma unroll
  for (int p = 0; p < 14; ++p) {
    v2f pr;
    pr.x = vals[2 * p];
    pr.y = vals[2 * p + 1];
    PB[p * P_TOTAL + g] = pr;
  }
  float4 pnc;
  pnc.x = nrm;
  pnc.y = vals[0 * 9 + 4];  // center c=0 (i=j=1)
  pnc.z = vals[1 * 9 + 4];  // center c=1
  pnc.w = vals[2 * 9 + 4];  // center c=2
  PNC[g] = pnc;
}

// ---------------------------------------------------------------------------
// Fused GEMM (WMMA f32 16x16x4, K padded to 28 -> 7 wmma per tile) with
// flash-style per-row online softmax epilogue. One block = one 16-row M tile;
// 4 waves split the 1800 N tiles; partials merged via shuffles + LDS.
// Per tile VMEM: 7x global_load_b64 (B pairs) + 1x global_load_b128 (PNC).
// ---------------------------------------------------------------------------
__global__ __launch_bounds__(128) void fused_score_kernel(
    const float* __restrict__ t,
    const float* __restrict__ x,
    const v2f* __restrict__ XA,
    const float* __restrict__ xnorms,
    const v2f* __restrict__ PB,
    const float4* __restrict__ PNC,
    float* __restrict__ out) {
  const int lane = threadIdx.x & 31;
  const int wave = threadIdx.x >> 5;     // 0..3
  const int hi = lane >> 4;              // K half (0/1) within 4-wide K group
  const int col = lane & 15;             // N column / M row-within-tile
  const int mbase = blockIdx.x * 16;

  // scalars from t
  const float tv = t[0];
  const float ang = tv * (float)(M_PI * 0.5 / 1.008);
  const float cc = __cosf(ang);
  const float bt2 = 1.0f - cc * cc;
  const float at = sqrtf(1.0f - bt2);
  const float inv2bt2 = 1.0f / (2.0f * bt2);
  const float cA = at / bt2;              // 2*at * inv2bt2
  const float cP = at * at * inv2bt2;     // at^2 / (2 bt^2)

  // A operands: lane holds row M = mbase+col, K pair p = 2q+hi
  v2f A[7];
#pragma unroll
  for (int q = 0; q < 7; ++q) {
    A[q] = XA[(2 * q + hi) * M_TOTAL + mbase + col];
  }

  // pre-scaled xnorm terms for this lane's 8 C-tile rows (M = mbase + r + 8*hi)
  float xnt[8];
#pragma unroll
  for (int r = 0; r < 8; ++r) xnt[r] = xnorms[mbase + r + 8 * hi] * inv2bt2;

  const float NEG_INF = -__builtin_huge_valf();
  float m_[8], s_[8], w0[8], w1[8], w2[8];
#pragma unroll
  for (int r = 0; r < 8; ++r) {
    m_[r] = NEG_INF; s_[r] = 0.0f; w0[r] = 0.0f; w1[r] = 0.0f; w2[r] = 0.0f;
  }

  const int t0 = wave * TILES_PER_WAVE;
  const int t1 = t0 + TILES_PER_WAVE;

  for (int nt = t0; nt < t1; ++nt) {
    const int ncol = nt * 16 + col;
    v8f Cacc = {0.0f, 0.0f, 0.0f, 0.0f, 0.0f, 0.0f, 0.0f, 0.0f};
#pragma unroll
    for (int q = 0; q < 7; ++q) {
      v2f Bv = PB[(2 * q + hi) * P_TOTAL + ncol];
      Cacc = __builtin_amdgcn_wmma_f32_16x16x4_f32(
          /*neg_a=*/false, A[q], /*neg_b=*/false, Bv,
          /*c_mod=*/(short)0, Cacc, /*reuse_a=*/false, /*reuse_b=*/false);
    }
    const float4 pnc = PNC[ncol];
    const float base = pnc.x * cP;  // at^2 * pnorm / (2 bt^2)
#pragma unroll
    for (int r = 0; r < 8; ++r) {
      float arg = fmaf(Cacc[r], cA, -(xnt[r] + base));
      float nm = fmaxf(m_[r], arg);
      float scale = __expf(m_[r] - nm);
      float e = __expf(arg - nm);
      m_[r] = nm;
      s_[r] = s_[r] * scale + e;
      w0[r] = w0[r] * scale + e * pnc.y;
      w1[r] = w1[r] * scale + e * pnc.z;
      w2[r] = w2[r] * scale + e * pnc.w;
    }
  }

  // merge softmax partials across the 16 N-columns (stays within half-wave)
#pragma unroll
  for (int mask = 1; mask < 16; mask <<= 1) {
#pragma unroll
    for (int r = 0; r < 8; ++r) {
      float om = __shfl_xor(m_[r], mask, 32);
      float os = __shfl_xor(s_[r], mask, 32);
      float o0 = __shfl_xor(w0[r], mask, 32);
      float o1 = __shfl_xor(w1[r], mask, 32);
      float o2 = __shfl_xor(w2[r], mask, 32);
      float nm = fmaxf(m_[r], om);
      float sa = __expf(m_[r] - nm);
      float sb = __expf(om - nm);
      s_[r] = s_[r] * sa + os * sb;
      w0[r] = w0[r] * sa + o0 * sb;
      w1[r] = w1[r] * sa + o1 * sb;
      w2[r] = w2[r] * sa + o2 * sb;
      m_[r] = nm;
    }
  }

  __shared__ float lds[WAVES][16][5];
  if (col == 0) {
#pragma unroll
    for (int r = 0; r < 8; ++r) {
      int row = r + 8 * hi;
      lds[wave][row][0] = m_[r];
      lds[wave][row][1] = s_[r];
      lds[wave][row][2] = w0[r];
      lds[wave][row][3] = w1[r];
      lds[wave][row][4] = w2[r];
    }
  }
  __syncthreads();

  if (threadIdx.x < 16) {
    int row = threadIdx.x;
    float m = NEG_INF, s = 0.0f, W0 = 0.0f, W1 = 0.0f, W2 = 0.0f;
#pragma unroll
    for (int wv = 0; wv < WAVES; ++wv) {
      float mm = lds[wv][row][0];
      float ss = lds[wv][row][1];
      float a0 = lds[wv][row][2];
      float a1 = lds[wv][row][3];
      float a2 = lds[wv][row][4];
      float nm = fmaxf(m, mm);
      float sa = __expf(m - nm);
      float sb = __expf(mm - nm);
      s = s * sa + ss * sb;
      W0 = W0 * sa + a0 * sb;
      W1 = W1 * sa + a1 * sb;
      W2 = W2 * sa + a2 * sb;
      m = nm;
    }
    int Mi = mbase + row;
    int b = Mi >> 10;
    int hw = Mi & (HW - 1);
    float invS = 1.0f / s;
    float ibt2 = 1.0f / bt2;
    float Wc[3] = {W0, W1, W2};
#pragma unroll
    for (int c = 0; c < C_N; ++c) {
      float xc = x[(b * C_N + c) * HW + hw];
      out[(b * C_N + c) * HW + hw] = (at * Wc[c] * invS - xc) * ibt2;
    }
  }
}

// ---------------------------------------------------------------------------
extern "C" void kernel_launch(void* const* d_in, const int* in_sizes, int n_in,
                              void* d_out, int out_size, void* d_ws, size_t ws_size,
                              hipStream_t stream) {
  const float* t = (const float*)d_in[0];
  const float* x = (const float*)d_in[1];
  const float* images = (const float*)d_in[2];
  float* out = (float*)d_out;

  float* ws = (float*)d_ws;
  v2f* XA = (v2f*)ws;                               // 14 * M_TOTAL v2f (448 KB)
  float* xnorms = ws + 2 * 14 * M_TOTAL;            // M_TOTAL floats
  v2f* PB = (v2f*)(xnorms + M_TOTAL);               // 14 * P_TOTAL v2f (3.2 MB)
  float4* PNC = (float4*)(xnorms + M_TOTAL + 2 * 14 * P_TOTAL);  // P_TOTAL float4

  build_x_kernel<<<(M_TOTAL + 255) / 256, 256, 0, stream>>>(x, XA, xnorms);
  build_p_kernel<<<(P_TOTAL + 255) / 256, 256, 0, stream>>>(images, PB, PNC);
  fused_score_kernel<<<M_TOTAL / 16, 128, 0, stream>>>(
      t, x, XA, xnorms, PB, PNC, out);
}